// PillarLayer_2989297238033
// MI455X (gfx1250) — compile-verified
//
#include <hip/hip_runtime.h>
#include <hip/hip_bf16.h>
#include <climits>

// ---- problem constants (from reference) ----
#define BATCH    8
#define NPT      150000        // points per batch
#define NXG      432
#define NYG      496
#define NZG      1
#define NCELLS   (NXG * NYG)   // 214272 (NZ==1)
#define MAXPTS   32
#define MAXVOX   16000
#define NBLK     ((NPT + 255) / 256)   // 586 point-blocks per batch
#define NBPAD    1024                  // padded stride for block sums

#define PC_LO_X  0.0f
#define PC_LO_Y  (-39.68f)
#define PC_LO_Z  (-3.0f)
#define VS_X     0.16f
#define VS_Y     0.16f
#define VS_Z     4.0f

// output layout (floats)
#define PIL_ELEMS   ((size_t)BATCH * MAXVOX * MAXPTS * 4)   // 16,384,000
#define COORS_BASE  PIL_ELEMS
#define COORS4_BASE (COORS_BASE / 4)                        // float4 index
#define NUM_BASE    (COORS_BASE + (size_t)BATCH * MAXVOX * 4)

// ---------------- K0: init workspace tables ----------------
__global__ void k0_init(int* __restrict__ grid_first, int* __restrict__ cell_slot,
                        int* __restrict__ vox_cell, int* __restrict__ fill) {
    int t = blockIdx.x * blockDim.x + threadIdx.x;
    if (t < BATCH * NCELLS) { grid_first[t] = INT_MAX; cell_slot[t] = -1; }
    if (t < BATCH * MAXVOX) { vox_cell[t] = -1; fill[t] = 0; }
}

// ---------------- K1: quantize points, first-index per cell ----------------
__global__ void k1_quant(const float4* __restrict__ pts, int* __restrict__ keys,
                         int* __restrict__ grid_first) {
    int t = blockIdx.x * blockDim.x + threadIdx.x;
    if (t >= BATCH * NPT) return;
    // stream-prefetch ahead (global_prefetch_b8)
    if ((t & 63) == 0) __builtin_prefetch(pts + t + 1024, 0, 0);
    int b = t / NPT;
    int i = t - b * NPT;
    float4 p = pts[t];
    float fx = floorf((p.x - PC_LO_X) / VS_X);
    float fy = floorf((p.y - PC_LO_Y) / VS_Y);
    float fz = floorf((p.z - PC_LO_Z) / VS_Z);
    bool valid = (fx >= 0.0f) & (fx < (float)NXG) &
                 (fy >= 0.0f) & (fy < (float)NYG) &
                 (fz >= 0.0f) & (fz < (float)NZG);
    int key = NCELLS;
    if (valid) {
        int cx = (int)fx, cy = (int)fy, cz = (int)fz;
        key = cz * (NYG * NXG) + cy * NXG + cx;
        atomicMin(&grid_first[b * NCELLS + key], i);
    }
    keys[t] = key;
}

// flag: this point is the first (smallest-index) point of its cell
__device__ __forceinline__ bool cell_first_flag(const int* keys, const int* grid_first,
                                                int b, int i, int& k) {
    k = NCELLS;
    if (i >= NPT) return false;
    k = keys[b * NPT + i];
    if (k >= NCELLS) return false;
    return grid_first[b * NCELLS + k] == i;
}

// ---------------- K2a: per-block count of cell-first flags ----------------
__global__ void k2a_blocksum(const int* __restrict__ keys, const int* __restrict__ grid_first,
                             int* __restrict__ bs) {
    int b = blockIdx.y;
    int tid = threadIdx.x;
    int i = blockIdx.x * 256 + tid;
    int k;
    bool flag = cell_first_flag(keys, grid_first, b, i, k);
    unsigned mask = (unsigned)__ballot((int)flag);
    int lane = tid & 31, wid = tid >> 5;
    __shared__ int wsum[8];
    if (lane == 0) wsum[wid] = __popc(mask);
    __syncthreads();
    if (tid == 0) {
        int s = 0;
        #pragma unroll
        for (int w = 0; w < 8; ++w) s += wsum[w];
        bs[b * NBPAD + blockIdx.x] = s;
    }
}

// ---------------- K2b: per-batch exclusive scan of block sums ----------------
__global__ void k2b_scan(int* __restrict__ bs) {
    int b = blockIdx.x;
    int tid = threadIdx.x;       // 1024 threads
    __shared__ int s[1024];
    int v = (tid < NBLK) ? bs[b * NBPAD + tid] : 0;
    s[tid] = v;
    __syncthreads();
    for (int off = 1; off < 1024; off <<= 1) {
        int t = (tid >= off) ? s[tid - off] : 0;
        __syncthreads();
        s[tid] += t;
        __syncthreads();
    }
    if (tid < NBLK) bs[b * NBPAD + tid] = s[tid] - v;   // exclusive
}

// ---------------- K3: assign slots (rank by first point index) ----------------
__global__ void k3_rank(const int* __restrict__ keys, const int* __restrict__ grid_first,
                        const int* __restrict__ bs, int* __restrict__ cell_slot,
                        int* __restrict__ vox_cell) {
    int b = blockIdx.y;
    int tid = threadIdx.x;
    int i = blockIdx.x * 256 + tid;
    int k;
    bool flag = cell_first_flag(keys, grid_first, b, i, k);
    unsigned mask = (unsigned)__ballot((int)flag);
    int lane = tid & 31, wid = tid >> 5;
    __shared__ int wsum[8], wbase[8];
    if (lane == 0) wsum[wid] = __popc(mask);
    __syncthreads();
    if (tid == 0) {
        int r = 0;
        #pragma unroll
        for (int w = 0; w < 8; ++w) { wbase[w] = r; r += wsum[w]; }
    }
    __syncthreads();
    if (flag) {
        int intra = wbase[wid] + __popc(mask & ((1u << lane) - 1u));
        int r = bs[b * NBPAD + blockIdx.x] + intra;
        if (r < MAXVOX) {
            cell_slot[b * NCELLS + k] = r;
            vox_cell[b * MAXVOX + r] = k;
        }
    }
}

// ---------------- K4: append point indices into pillars ----------------
__global__ void k4_append(const int* __restrict__ keys, const int* __restrict__ cell_slot,
                          int* __restrict__ fill, int* __restrict__ pidx) {
    int t = blockIdx.x * blockDim.x + threadIdx.x;
    if (t >= BATCH * NPT) return;
    int b = t / NPT;
    int i = t - b * NPT;
    int k = keys[t];
    if (k >= NCELLS) return;
    int s = cell_slot[b * NCELLS + k];
    if (s < 0) return;
    int g = b * MAXVOX + s;
    int pos = atomicAdd(&fill[g], 1);
    if (pos < MAXPTS) pidx[(size_t)g * MAXPTS + pos] = i;
}

// ---------------- K5: per-pillar wave32 bitonic sort + gather + emit ----------------
__global__ void k5_finalize(const float4* __restrict__ pts, const int* __restrict__ vox_cell,
                            const int* __restrict__ fill, const int* __restrict__ pidx,
                            float* __restrict__ out) {
    int tid = threadIdx.x;
    int lane = tid & 31;
    int g = blockIdx.x * 8 + (tid >> 5);      // 8 pillars (waves) per block; 16000*8 = 128000 exact
    int b = g / MAXVOX;
    int cell = vox_cell[g];
    int cnt = fill[g];
    int m = (cell >= 0) ? ((cnt < MAXPTS) ? cnt : MAXPTS) : 0;
    int idx = (lane < m) ? pidx[(size_t)g * MAXPTS + lane] : INT_MAX;
    // in-wave bitonic sort over 32 lanes (restores deterministic point-index order)
    #pragma unroll
    for (int k = 2; k <= 32; k <<= 1) {
        #pragma unroll
        for (int j = k >> 1; j > 0; j >>= 1) {
            int other = __shfl_xor(idx, j, 32);
            bool tmin = ((lane & k) == 0) == ((lane & j) == 0);
            idx = tmin ? (idx < other ? idx : other) : (idx > other ? idx : other);
        }
    }
    float4 v = make_float4(0.f, 0.f, 0.f, 0.f);
    if (idx != INT_MAX) v = pts[(size_t)b * NPT + idx];
    ((float4*)out)[(size_t)g * MAXPTS + lane] = v;    // zeros for empty entries
    if (lane == 0) {
        float4 cb;
        if (cell >= 0) {
            int x = cell % NXG;
            int y = (cell / NXG) % NYG;
            int z = cell / (NXG * NYG);
            cb = make_float4((float)z, (float)y, (float)x, (float)b);
        } else {
            cb = make_float4(-1.f, -1.f, -1.f, (float)b);
        }
        ((float4*)out)[COORS4_BASE + g] = cb;
        out[NUM_BASE + g] = (float)m;
    }
}

extern "C" void kernel_launch(void* const* d_in, const int* in_sizes, int n_in,
                              void* d_out, int out_size, void* d_ws, size_t ws_size,
                              hipStream_t stream) {
    const float4* pts = (const float4*)d_in[0];   // (8, 150000, 4) float32
    float* out = (float*)d_out;

    // workspace layout (ints): ~36 MB total
    int* ws         = (int*)d_ws;
    int* keys       = ws;                                  // B*NPT
    int* grid_first = keys + (size_t)BATCH * NPT;          // B*NCELLS
    int* cell_slot  = grid_first + (size_t)BATCH * NCELLS; // B*NCELLS
    int* vox_cell   = cell_slot + (size_t)BATCH * NCELLS;  // B*MAXVOX
    int* fill       = vox_cell + (size_t)BATCH * MAXVOX;   // B*MAXVOX
    int* bs         = fill + (size_t)BATCH * MAXVOX;       // B*NBPAD
    int* pidx       = bs + (size_t)BATCH * NBPAD;          // B*MAXVOX*MAXPTS

    k0_init<<<(BATCH * NCELLS + 255) / 256, 256, 0, stream>>>(grid_first, cell_slot, vox_cell, fill);
    k1_quant<<<(BATCH * NPT + 255) / 256, 256, 0, stream>>>(pts, keys, grid_first);
    k2a_blocksum<<<dim3(NBLK, BATCH), 256, 0, stream>>>(keys, grid_first, bs);
    k2b_scan<<<BATCH, 1024, 0, stream>>>(bs);
    k3_rank<<<dim3(NBLK, BATCH), 256, 0, stream>>>(keys, grid_first, bs, cell_slot, vox_cell);
    k4_append<<<(BATCH * NPT + 255) / 256, 256, 0, stream>>>(keys, cell_slot, fill, pidx);
    k5_finalize<<<(BATCH * MAXVOX) / 8, 256, 0, stream>>>(pts, vox_cell, fill, pidx, out);
}